// vector_quantizer_4990751998021
// MI455X (gfx1250) — compile-verified
//
#include <hip/hip_runtime.h>
#include <math.h>

#define DIM     20
#define LSEQ    4096
#define NBATCH  32
#define KCODES  1000
#define NROWS   (NBATCH * LSEQ)          /* 131072 */
#define NCHUNK  63                        /* ceil(1000/16) */
#define QELEMS  (NBATCH * DIM * LSEQ)     /* 2621440 */
#define FLT_BIG 3.402823466e38f

typedef float v2f __attribute__((ext_vector_type(2)));
typedef float v8f __attribute__((ext_vector_type(8)));

// ---------------------------------------------------------------- init ----
__global__ void vq_init_kernel(unsigned int* __restrict__ counts,
                               float* __restrict__ klp) {
    int t = blockIdx.x * blockDim.x + threadIdx.x;   // 1024 threads total
    if (t < 1024) counts[t] = 0u;
    if (t < 1024) klp[t]    = 0.0f;
}

// ---------------------------------------------------------------- main ----
__global__ __launch_bounds__(256) void vq_main_kernel(
    const float* __restrict__ x,       // [B, D, L]
    const float* __restrict__ emb,     // [K, D]
    float* __restrict__ outq,          // [B, D, L]
    unsigned int* __restrict__ counts, // [1024] (first 1000 used)
    float* __restrict__ klp)           // [gridDim.x]
{
    __shared__ float s_e2[1008];
    __shared__ float s_xt[8][16][DIM];
    __shared__ int   s_idx[8][16];
    __shared__ float s_kl[8];

    const int tid  = threadIdx.x;
    const int wave = tid >> 5;
    const int lane = tid & 31;
    const int hl   = lane & 15;          // position within half-wave
    const int koff = (lane >> 4) * 2;    // K split across lane halves

    // --- ||e||^2 for all codes (pad ragged tail with +BIG so it never wins)
    for (int j = tid; j < 1008; j += 256) {
        float s = FLT_BIG;
        if (j < KCODES) {
            s = 0.0f;
            const float* er = emb + j * DIM;
            #pragma unroll
            for (int d = 0; d < DIM; ++d) { float v = er[d]; s = fmaf(v, v, s); }
        }
        s_e2[j] = s;
    }

    // --- stage x tile (128 consecutive rows, same b) into LDS, coalesced
    const int rowbase = blockIdx.x * 128;
    const int b  = rowbase >> 12;            // L = 4096
    const int l0 = rowbase & (LSEQ - 1);
    for (int e = tid; e < 128 * DIM; e += 256) {
        int d = e >> 7;                      // 0..19
        int r = e & 127;                     // 0..127 -> consecutive l
        s_xt[r >> 4][r & 15][d] =
            x[(size_t)b * DIM * LSEQ + (size_t)d * LSEQ + (size_t)(l0 + r)];
    }
    __syncthreads();

    // --- A fragments (16x4 f32 WMMA layout): lane m = hl, V0=K(koff), V1=K(koff+1)
    v2f afrag[5];
    const float* xr = &s_xt[wave][hl][0];
    #pragma unroll
    for (int kc = 0; kc < 5; ++kc)
        afrag[kc] = *(const v2f*)(xr + kc * 4 + koff);   // ds_load_2addr_b64

    float bestv[8];
    int   besti[8];
    #pragma unroll
    for (int r = 0; r < 8; ++r) { bestv[r] = FLT_BIG; besti[r] = 0; }

    // --- fused GEMM + argmin over 63 chunks of 16 codes
    for (int c = 0; c < NCHUNK; ++c) {
        const int col  = c * 16 + hl;
        // Clamp instead of predicating: out-of-range lanes read row 999's data,
        // but s_e2[col] = FLT_BIG guarantees they never win the argmin.
        // This keeps the hot loop free of exec-mask manipulation.
        const int colc = (col < KCODES) ? col : (KCODES - 1);

        v2f bfrag[5];
        const float* ebase = emb + (size_t)colc * DIM + koff;
        #pragma unroll
        for (int kc = 0; kc < 5; ++kc)
            bfrag[kc] = *(const v2f*)(ebase + kc * 4);   // global_load_b64, L2-hot

        v8f acc = {};
        #pragma unroll
        for (int kc = 0; kc < 5; ++kc)
            acc = __builtin_amdgcn_wmma_f32_16x16x4_f32(
                      false, afrag[kc], false, bfrag[kc],
                      (short)0, acc, false, false);

        const float e2c = s_e2[c * 16 + hl];
        #pragma unroll
        for (int r = 0; r < 8; ++r) {
            float dv = fmaf(-2.0f, acc[r], e2c);   // ||x||^2 constant per row
            if (dv < bestv[r]) { bestv[r] = dv; besti[r] = col; }
        }
    }

    // --- per-row argmin: reduce across the 16 lanes sharing a row
    #pragma unroll
    for (int r = 0; r < 8; ++r) {
        #pragma unroll
        for (int m = 1; m < 16; m <<= 1) {
            float ov = __shfl_xor(bestv[r], m, 16);
            int   oi = __shfl_xor(besti[r], m, 16);
            if (ov < bestv[r] || (ov == bestv[r] && oi < besti[r])) {
                bestv[r] = ov; besti[r] = oi;
            }
        }
        // lanes 0-15 hold row r, lanes 16-31 hold row r+8
        if (hl == 0) s_idx[wave][r + (lane >> 4) * 8] = besti[r];
    }
    __syncthreads();

    // --- epilogue: gather code, write output, histogram, commitment term
    float kl = 0.0f;
    if (lane < 16) {
        const int m   = lane;
        const int n   = rowbase + wave * 16 + m;
        const int bb  = n >> 12;
        const int ll  = n & (LSEQ - 1);
        const int idx = s_idx[wave][m];

        atomicAdd(&counts[idx], 1u);   // integer: associative, deterministic

        float e_row[DIM], x_row[DIM];
        const float* er = emb + (size_t)idx * DIM;
        #pragma unroll
        for (int d = 0; d < DIM; ++d) e_row[d] = er[d];
        #pragma unroll
        for (int d = 0; d < DIM; ++d) x_row[d] = s_xt[wave][m][d];

        // quantized output, [B, D, L]; coalesced across lanes for each d
        #pragma unroll
        for (int d = 0; d < DIM; ++d)
            outq[(size_t)bb * DIM * LSEQ + (size_t)d * LSEQ + (size_t)ll] = e_row[d];

        // t = softmax(x_row), p = softmax(e_row); kl = sum t*(log t - p)
        float mx = x_row[0], me = e_row[0];
        #pragma unroll
        for (int d = 1; d < DIM; ++d) { mx = fmaxf(mx, x_row[d]); me = fmaxf(me, e_row[d]); }
        float sx = 0.0f, se = 0.0f;
        #pragma unroll
        for (int d = 0; d < DIM; ++d) { sx += __expf(x_row[d] - mx); se += __expf(e_row[d] - me); }
        const float lse_x  = __logf(sx);
        const float inv_sx = 1.0f / sx;
        const float inv_se = 1.0f / se;
        #pragma unroll
        for (int d = 0; d < DIM; ++d) {
            float ex   = __expf(x_row[d] - mx);
            float t    = ex * inv_sx;
            float logt = (x_row[d] - mx) - lse_x;
            float p    = __expf(e_row[d] - me) * inv_se;
            kl = fmaf(t, logt - p, kl);
        }
    }
    // fixed-order wave reduction of kl (deterministic)
    #pragma unroll
    for (int m = 16; m >= 1; m >>= 1) kl += __shfl_xor(kl, m, 32);
    if (lane == 0) s_kl[wave] = kl;
    __syncthreads();
    if (tid == 0) {
        float s = 0.0f;
        #pragma unroll
        for (int w = 0; w < 8; ++w) s += s_kl[w];
        klp[blockIdx.x] = s;
    }
}

// ------------------------------------------------------------- finalize ----
__global__ void vq_final_kernel(const unsigned int* __restrict__ counts,
                                const float* __restrict__ klp,
                                float* __restrict__ out_scalars) {
    __shared__ float red[256];
    const int t = threadIdx.x;

    // commitment-loss sum over 1024 block partials (fixed order)
    float s = 0.0f;
    for (int i = t; i < 1024; i += 256) s += klp[i];
    red[t] = s; __syncthreads();
    for (int off = 128; off > 0; off >>= 1) {
        if (t < off) red[t] += red[t + off];
        __syncthreads();
    }
    const float klsum = red[0];
    __syncthreads();

    // entropy of code usage (fixed order)
    float h = 0.0f;
    for (int j = t; j < KCODES; j += 256) {
        float avg = (float)counts[j] * (1.0f / (float)NROWS);
        h += avg * logf(avg + 1e-10f);
    }
    red[t] = h; __syncthreads();
    for (int off = 128; off > 0; off >>= 1) {
        if (t < off) red[t] += red[t + off];
        __syncthreads();
    }
    if (t == 0) {
        out_scalars[0] = 0.1f * (klsum / (float)NBATCH);  // loss
        out_scalars[1] = expf(-red[0]);                   // perplexity
    }
}

// --------------------------------------------------------------- launch ----
extern "C" void kernel_launch(void* const* d_in, const int* in_sizes, int n_in,
                              void* d_out, int out_size, void* d_ws, size_t ws_size,
                              hipStream_t stream) {
    (void)in_sizes; (void)n_in; (void)out_size; (void)ws_size;

    const float* x   = (const float*)d_in[0];   // [B, D, L]
    const float* emb = (const float*)d_in[1];   // [K, D]
    float* out = (float*)d_out;

    unsigned int* counts = (unsigned int*)d_ws;                  // 4 KB
    float*        klp    = (float*)((char*)d_ws + 4096);         // 4 KB

    vq_init_kernel<<<4, 256, 0, stream>>>(counts, klp);
    vq_main_kernel<<<NROWS / 128, 256, 0, stream>>>(x, emb, out, counts, klp);
    vq_final_kernel<<<1, 256, 0, stream>>>(counts, klp, out + QELEMS);
}